// Criterion_29386166239267
// MI455X (gfx1250) — compile-verified
//
#include <hip/hip_runtime.h>
#include <math.h>

typedef __attribute__((ext_vector_type(2))) float v2f;
typedef __attribute__((ext_vector_type(8))) float v8f;

#define NT        2048
#define BS        4096
#define ROW       ((BS - 1) * NT)      // only the last row survives in the reference
#define NTHREADS  256
#define NWAVES    (NTHREADS / 32)
#define LOG_2PI   1.8378770664093453f

// Exact f32 wave-wide sum using V_WMMA_F32_16X16X4_F32.
// A: each lane holds {partial, 0} (16x4 f32 A = 2 VGPRs/lane), B = all ones.
// D[m,n] = sum_k A[m,k]  ->  sum over D rows (8 VGPR adds + 1 shfl_xor(16))
// gives the full 32-lane sum in every lane, layout-independently.
__device__ __forceinline__ float wave_sum_wmma(float partial) {
    v2f a; a[0] = partial; a[1] = 0.0f;
    v2f b; b[0] = 1.0f;    b[1] = 1.0f;
    v8f c = {};
    c = __builtin_amdgcn_wmma_f32_16x16x4_f32(
        /*neg_a=*/false, a, /*neg_b=*/false, b,
        /*c_mod=*/(short)0, c, /*reuse_a=*/false, /*reuse_b=*/false);
    float t = ((c[0] + c[1]) + (c[2] + c[3])) + ((c[4] + c[5]) + (c[6] + c[7]));
    t += __shfl_xor(t, 16, 32);
    return t;
}

__device__ __forceinline__ float block_sum(float v, float* red, int lane, int wid) {
    v = wave_sum_wmma(v);
    __syncthreads();                    // protect `red` reuse from previous call
    if (lane == 0) red[wid] = v;
    __syncthreads();
    float r = 0.0f;
#pragma unroll
    for (int i = 0; i < NWAVES; ++i) r += red[i];
    return r;
}

__device__ __forceinline__ float block_max(float v, float* red, int lane, int wid) {
#pragma unroll
    for (int off = 16; off > 0; off >>= 1)
        v = fmaxf(v, __shfl_xor(v, off, 32));
    __syncthreads();
    if (lane == 0) red[wid] = v;
    __syncthreads();
    float r = red[0];
#pragma unroll
    for (int i = 1; i < NWAVES; ++i) r = fmaxf(r, red[i]);
    return r;
}

__global__ __launch_bounds__(NTHREADS)
void criterion_kernel(const float* __restrict__ prior_mu,
                      const float* __restrict__ prior_sigma,
                      const float* __restrict__ mu,
                      const float* __restrict__ sigma,
                      const float* __restrict__ target_y,
                      const float* __restrict__ eps_post,
                      const float* __restrict__ eps_prior,
                      float* __restrict__ out) {
    __shared__ float zq[NT];            // z_post  (posterior sample)
    __shared__ float zp[NT];            // z_prior (prior sample)
    __shared__ float red[NWAVES];

    const int tid  = threadIdx.x;
    const int lane = tid & 31;
    const int wid  = tid >> 5;

    // Speculative prefetch of the single live row (emits global_prefetch_b8).
    {
        const int off = ROW + tid * (NT / NTHREADS);
        __builtin_prefetch(&prior_mu[off],    0, 3);
        __builtin_prefetch(&prior_sigma[off], 0, 3);
        __builtin_prefetch(&mu[off],          0, 3);
        __builtin_prefetch(&sigma[off],       0, 3);
        __builtin_prefetch(&target_y[off],    0, 3);
        __builtin_prefetch(&eps_post[off],    0, 3);
        __builtin_prefetch(&eps_prior[off],   0, 3);
    }

    // ---- Pass 1: load row, build z's in LDS, accumulate single-pass sums ----
    float maxq = -__builtin_inff(), maxp = -__builtin_inff();
    float s_logsig = 0.0f, s_quad = 0.0f;

    const float4* mu4 = (const float4*)(mu          + ROW);
    const float4* sg4 = (const float4*)(sigma       + ROW);
    const float4* ep4 = (const float4*)(eps_post    + ROW);
    const float4* pm4 = (const float4*)(prior_mu    + ROW);
    const float4* ps4 = (const float4*)(prior_sigma + ROW);
    const float4* pe4 = (const float4*)(eps_prior   + ROW);
    const float4* ty4 = (const float4*)(target_y    + ROW);

    auto do_elem = [&](int j, float m, float s, float e,
                       float pm, float ps, float pe, float y) {
        float zpost = m  + sqrtf(s)  * e;
        float zpri  = pm + sqrtf(ps) * pe;
        zq[j] = zpost;
        zp[j] = zpri;
        maxq = fmaxf(maxq, zpost);
        maxp = fmaxf(maxp, zpri);
        s_logsig += logf(s);
        float d = y - m;
        s_quad += d * d / s;
    };

#pragma unroll
    for (int v = tid; v < NT / 4; v += NTHREADS) {
        float4 m  = mu4[v], s  = sg4[v], e  = ep4[v];
        float4 pm = pm4[v], ps = ps4[v], pe = pe4[v];
        float4 y  = ty4[v];
        int j = v * 4;
        do_elem(j + 0, m.x, s.x, e.x, pm.x, ps.x, pe.x, y.x);
        do_elem(j + 1, m.y, s.y, e.y, pm.y, ps.y, pe.y, y.y);
        do_elem(j + 2, m.z, s.z, e.z, pm.z, ps.z, pe.z, y.z);
        do_elem(j + 3, m.w, s.w, e.w, pm.w, ps.w, pe.w, y.w);
    }

    maxq = block_max(maxq, red, lane, wid);
    maxp = block_max(maxp, red, lane, wid);
    float S_logsig = block_sum(s_logsig, red, lane, wid);
    float S_quad   = block_sum(s_quad,   red, lane, wid);

    // ---- Pass 2: log-sum-exp denominators from LDS ----
    float seq = 0.0f, sep = 0.0f;
    for (int j = tid; j < NT; j += NTHREADS) {
        seq += expf(zq[j] - maxq);
        sep += expf(zp[j] - maxp);
    }
    float SEQ = block_sum(seq, red, lane, wid);
    float SEP = block_sum(sep, red, lane, wid);
    float Lq = maxq + logf(SEQ);        // logsumexp(z_post)
    float Lp = maxp + logf(SEP);        // logsumexp(z_prior)

    // ---- Pass 3: KL term  sum p * (log_p - log_q) ----
    float skl = 0.0f;
    for (int j = tid; j < NT; j += NTHREADS) {
        float lp = zp[j] - Lp;          // log_p
        float lq = zq[j] - Lq;          // log_q
        skl += expf(lp) * (lp - lq);
    }
    float SKL = block_sum(skl, red, lane, wid);

    if (tid == 0) {
        float kl = SKL / (float)NT;     // mean over last axis
        float log_prob = -0.5f * ((float)NT * LOG_2PI + S_logsig + S_quad);
        out[0] = -(log_prob / (float)NT - kl);
    }
}

extern "C" void kernel_launch(void* const* d_in, const int* in_sizes, int n_in,
                              void* d_out, int out_size, void* d_ws, size_t ws_size,
                              hipStream_t stream) {
    (void)in_sizes; (void)n_in; (void)out_size; (void)d_ws; (void)ws_size;
    criterion_kernel<<<1, NTHREADS, 0, stream>>>(
        (const float*)d_in[0],   // prior_mu
        (const float*)d_in[1],   // prior_sigma
        (const float*)d_in[2],   // mu
        (const float*)d_in[3],   // sigma
        (const float*)d_in[4],   // target_y
        (const float*)d_in[5],   // eps_post
        (const float*)d_in[6],   // eps_prior
        (float*)d_out);
}